// Repro_63428077027474
// MI455X (gfx1250) — compile-verified
//
#include <hip/hip_runtime.h>
#include <stdint.h>

// ---------------------------------------------------------------------------
// GCN edge normalization: deg = scatter_add(w, col); dis = rsqrt0(deg);
// out[e] = dis[row[e]] * dis[col[e]]
// Memory-bound: ~128 MB HBM traffic -> ~5.5us floor at 23.3 TB/s.
// deg/dis table (400KB) and the col stream stay L2-resident (192MB L2);
// w load and out store are non-temporal to avoid polluting L2.
// ---------------------------------------------------------------------------

static constexpr int kNodes = 100000;

// native vector type for __builtin_nontemporal_* (HIP_vector_type is rejected)
typedef float __attribute__((ext_vector_type(4))) v4f;

// ---- CDNA5 async global->LDS copy (ASYNCcnt path), inline asm -------------
__device__ __forceinline__ void async_copy_b128(unsigned lds_byte_off, const void* gaddr) {
  // LDS byte address comes from lds_byte_off; 16 bytes per lane.
  asm volatile("global_load_async_to_lds_b128 %0, %1, off"
               :: "v"(lds_byte_off), "v"(gaddr)
               : "memory");
}

__device__ __forceinline__ void atomic_add_f32(float* p, float v) {
  // native global_atomic_add_f32 without return (STOREcnt, no CAS loop)
  unsafeAtomicAdd(p, v);
}

// ---- kernel 0: zero degree array ------------------------------------------
__global__ void zero_f32(float* __restrict__ p, int n) {
  int i = blockIdx.x * blockDim.x + threadIdx.x;
  if (i < n) p[i] = 0.0f;
}

// ---- kernel 1: degree scatter (vectorized, grid-stride, guarded prefetch) -
__global__ __launch_bounds__(256)
void degree_scatter(const int* __restrict__ col, const float* __restrict__ w,
                    float* __restrict__ deg, int e4) {
  const int4* c4 = reinterpret_cast<const int4*>(col);
  const v4f*  w4 = reinterpret_cast<const v4f*>(w);
  int stride = gridDim.x * blockDim.x;
  for (int i = blockIdx.x * blockDim.x + threadIdx.x; i < e4; i += stride) {
    int nxt = i + stride;
    if (nxt < e4) {
      // CDNA5 global_prefetch_b8 of next grid-stride iteration (in-bounds only)
      __builtin_prefetch((const void*)(c4 + nxt), 0, 0);
      __builtin_prefetch((const void*)(w4 + nxt), 0, 0);
    }
    int4 c = c4[i];
    v4f  v = __builtin_nontemporal_load(w4 + i);  // w is read exactly once
    atomic_add_f32(&deg[c.x], v.x);
    atomic_add_f32(&deg[c.y], v.y);
    atomic_add_f32(&deg[c.z], v.z);
    atomic_add_f32(&deg[c.w], v.w);
  }
}

__global__ void degree_scatter_scalar(const int* __restrict__ col,
                                      const float* __restrict__ w,
                                      float* __restrict__ deg, int start, int E) {
  int i = start + blockIdx.x * blockDim.x + threadIdx.x;
  if (i < E) atomic_add_f32(&deg[col[i]], w[i]);
}

// ---- kernel 2: in-place deg -> inf-safe rsqrt(deg) ------------------------
__global__ void deg_inv_sqrt(float* __restrict__ deg, int n) {
  int i = blockIdx.x * blockDim.x + threadIdx.x;
  if (i < n) {
    float d = deg[i];
    deg[i] = (d == 0.0f) ? 0.0f : rsqrtf(d);
  }
}

// ---- kernel 3: gather + multiply, async double-buffered index staging -----
// Per wave: tile of 128 edges. Each lane async-copies 16B of row and 16B of
// col into a wave-private LDS buffer; two buffers per wave pipeline tile t+1's
// fetch behind tile t's L2 gathers. s_wait_asynccnt tracks completion.
__global__ __launch_bounds__(256)
void edge_norm_async(const int* __restrict__ row, const int* __restrict__ col,
                     const float* __restrict__ dis, float* __restrict__ out,
                     int ntiles) {
  __shared__ int smem[4096];  // 8 waves * 2 bufs * (128 row + 128 col) ints = 16KB

  const int lane = threadIdx.x & 31;
  const int wave = threadIdx.x >> 5;
  const unsigned ldsBase = (unsigned)(uintptr_t)(&smem[0]);
  const unsigned wb = ldsBase + (unsigned)wave * 2048u;  // bytes: 2 bufs * 1KB

  const int wavesPerBlock = blockDim.x >> 5;
  const int nwaves = (int)gridDim.x * wavesPerBlock;
  int t = (int)blockIdx.x * wavesPerBlock + wave;

  auto issue = [&](int tt, int b) {
    const int* gr = row + (size_t)tt * 128 + (size_t)lane * 4;
    const int* gc = col + (size_t)tt * 128 + (size_t)lane * 4;
    unsigned ro = wb + (unsigned)b * 1024u + (unsigned)lane * 16u;
    async_copy_b128(ro, gr);          // 128 row indices / wave
    async_copy_b128(ro + 512u, gc);   // 128 col indices / wave
  };

  int cur = 0;
  if (t < ntiles) issue(t, 0);

  for (; t < ntiles; t += nwaves) {
    int tn = t + nwaves;
    if (tn < ntiles) {
      issue(tn, cur ^ 1);
      // 4 async ops outstanding; first 2 (current tile, in-order) done at cnt<=2
      asm volatile("s_wait_asynccnt 2" ::: "memory");
    } else {
      asm volatile("s_wait_asynccnt 0" ::: "memory");
    }

    const int base = wave * 512 + cur * 256 + lane * 4;  // int index into smem
    int r0 = smem[base + 0],   r1 = smem[base + 1];
    int r2 = smem[base + 2],   r3 = smem[base + 3];
    int c0 = smem[base + 128], c1 = smem[base + 129];
    int c2 = smem[base + 130], c3 = smem[base + 131];

    v4f o;
    o.x = dis[r0] * dis[c0];
    o.y = dis[r1] * dis[c1];
    o.z = dis[r2] * dis[c2];
    o.w = dis[r3] * dis[c3];
    // out is write-once, never re-read: non-temporal store keeps L2 for
    // the dis table and the col stream.
    __builtin_nontemporal_store(o, reinterpret_cast<v4f*>(out) + (size_t)t * 32 + lane);

    cur ^= 1;
  }
}

__global__ void edge_norm_scalar(const int* __restrict__ row,
                                 const int* __restrict__ col,
                                 const float* __restrict__ dis,
                                 float* __restrict__ out, int start, int E) {
  int i = start + blockIdx.x * blockDim.x + threadIdx.x;
  if (i < E) out[i] = dis[row[i]] * dis[col[i]];
}

// ---------------------------------------------------------------------------
extern "C" void kernel_launch(void* const* d_in, const int* in_sizes, int n_in,
                              void* d_out, int out_size, void* d_ws, size_t ws_size,
                              hipStream_t stream) {
  const int*   edge_index = (const int*)d_in[0];   // [2, E] int32
  const float* w          = (const float*)d_in[1]; // [E] float32
  const int E = in_sizes[1];
  const int* row = edge_index;       // edge_index[0]
  const int* col = edge_index + E;   // edge_index[1]
  float* out = (float*)d_out;
  float* deg = (float*)d_ws;         // N floats, re-zeroed each call
  const int N = kNodes;

  zero_f32<<<(N + 255) / 256, 256, 0, stream>>>(deg, N);

  const bool vecOK = ((E & 3) == 0);  // int4/float4 + b128 alignment of col

  if (vecOK && E >= 4) {
    int e4 = E >> 2;
    int blocks = (e4 + 255) / 256;
    if (blocks > 2048) blocks = 2048;  // grid-stride persistent-ish
    degree_scatter<<<blocks, 256, 0, stream>>>(col, w, deg, e4);
  } else if (E > 0) {
    degree_scatter_scalar<<<(E + 255) / 256, 256, 0, stream>>>(col, w, deg, 0, E);
  }

  deg_inv_sqrt<<<(N + 255) / 256, 256, 0, stream>>>(deg, N);

  int done = 0;
  if (vecOK) {
    int ntiles = E / 128;  // 128 edges per wave-tile
    if (ntiles > 0) {
      int blocks = (ntiles + 7) / 8;           // 8 waves per block
      if (blocks > 2048) blocks = 2048;        // grid-stride over tiles
      edge_norm_async<<<blocks, 256, 0, stream>>>(row, col, deg, out, ntiles);
      done = ntiles * 128;
    }
  }
  if (done < E) {
    edge_norm_scalar<<<(E - done + 255) / 256, 256, 0, stream>>>(row, col, deg, out, done, E);
  }
}